// MPC_16690242912927
// MI455X (gfx1250) — compile-verified
//
#include <hip/hip_runtime.h>

// Batched LQR (Riccati backward + forward rollout), one wave32 per batch element.
// NT = 16 == WMMA tile size; Q = C + F^T V F done with v_wmma_f32_16x16x4_f32 chains.
// Next-step C/F tiles are software-prefetched (global_prefetch_b8) to hide the
// serial t-recursion's load latency; staging loads use b128.

static constexpr int TS = 128;   // time steps
static constexpr int BB = 2048;  // batch
static constexpr int NS = 12;    // states
static constexpr int NC = 4;     // controls
static constexpr int NT = 16;    // NS + NC
static constexpr int NW = 8;     // waves per block

typedef float v2f __attribute__((ext_vector_type(2)));
typedef float v8f __attribute__((ext_vector_type(8)));

__device__ __forceinline__ v8f v8zero() {
    v8f a;
#pragma unroll
    for (int r = 0; r < 8; ++r) a[r] = 0.f;
    return a;
}

// D = op(X) * Y + acc, where X,Y are 16x16 tiles in LDS (row-major, stride 17).
// xt=true uses X transposed. Chains 4x V_WMMA_F32_16X16X4_F32 along K.
__device__ __forceinline__ v8f mm16(const float* X, const float* Y, v8f acc,
                                    int m, int hi, bool xt) {
#pragma unroll
    for (int k4 = 0; k4 < 4; ++k4) {
        const int k0 = k4 * 4 + hi * 2;
        v2f a, b;
        if (xt) { a.x = X[k0 * 17 + m]; a.y = X[(k0 + 1) * 17 + m]; }
        else    { a.x = X[m * 17 + k0]; a.y = X[m * 17 + k0 + 1]; }
        b.x = Y[k0 * 17 + m]; b.y = Y[(k0 + 1) * 17 + m];
        acc = __builtin_amdgcn_wmma_f32_16x16x4_f32(false, a, false, b,
                                                    (short)0, acc, false, false);
    }
    return acc;
}

// Scatter WMMA C/D layout (8 VGPRs) into a 16x17 LDS tile.
__device__ __forceinline__ void storeCD(float* Z, v8f d, int n, int hi) {
    const int mb = hi * 8;
#pragma unroll
    for (int r = 0; r < 8; ++r) Z[(mb + r) * 17 + n] = d[r];
}

__global__ __launch_bounds__(NW * 32) void lqr_kernel(
    const float* __restrict__ Cg, const float* __restrict__ cg,
    const float* __restrict__ Fg, const float* __restrict__ fg,
    const float* __restrict__ x0g, float* __restrict__ xs,
    float* __restrict__ us, float* __restrict__ costs,
    float* __restrict__ Kk) {
    __shared__ float sV[NW][16][17];   // padded V (zeros outside 12x12)
    __shared__ float sF[NW][16][17];   // padded F (zeros in rows 12..15)
    __shared__ float sM[NW][16][17];   // M = V*F, later unsymmetrized Vn temp
    __shared__ float sQ[NW][16][17];   // Q tile; C staging in forward pass
    __shared__ float sKk[NW][64];      // K[4][12] at 0..47, k at 48..51
    __shared__ float sq[NW][16];       // q vector
    __shared__ float sW[NW][16];       // W = V f + v
    __shared__ float sv[NW][16];       // v vector
    __shared__ float sR[NW][16];       // f staging
    __shared__ float sx[NW][16];       // tau = [x; u] in forward pass

    const int lane = threadIdx.x & 31;
    const int w = threadIdx.x >> 5;
    const int b = blockIdx.x * NW + w;
    const int m = lane & 15;
    const int hi = (lane >> 4) & 1;

    float* V = &sV[w][0][0];
    float* F = &sF[w][0][0];
    float* M = &sM[w][0][0];
    float* Q = &sQ[w][0][0];
    float* KK = sKk[w];
    float* qv = sq[w];
    float* Wv = sW[w];
    float* vv = sv[w];
    float* fv = sR[w];
    float* xv = sx[w];

    // V = 0, F padding rows = 0, v = 0
    for (int i = lane; i < 16 * 17; i += 32) { V[i] = 0.f; F[i] = 0.f; }
    if (lane < 16) vv[lane] = 0.f;
    __syncthreads();

    // ---------------- Backward Riccati pass ----------------
    for (int t = TS - 1; t >= 0; --t) {
        const size_t tb = (size_t)t * BB + b;
        const float* Fp = Fg + tb * (NS * NT);
        const float* fp = fg + tb * NS;
        const float* Cp = Cg + tb * (NT * NT);
        const float* cp = cg + tb * NT;

        // Prefetch next step's C (1KB, lanes 0..7) and F (768B, lanes 8..13)
        // tiles while this step computes: hides latency on the serial chain.
        if (t > 0) {
            const char* pfa = (lane < 8)
                ? ((const char*)Cp - (size_t)BB * sizeof(float) * NT * NT +
                   (size_t)lane * 128)
                : ((const char*)Fp - (size_t)BB * sizeof(float) * NS * NT +
                   (size_t)(lane - 8) * 128);
            if (lane < 14) __builtin_prefetch(pfa, 0, 3);
        }

        {   // F tile: b128 loads, scalar LDS scatter (17-float pitch)
            const float4* Fp4 = (const float4*)Fp;
            for (int i = lane; i < (NS * NT) / 4; i += 32) {
                const float4 x = Fp4[i];
                float* dst = &F[(i / 4) * 17 + (i % 4) * 4];
                dst[0] = x.x; dst[1] = x.y; dst[2] = x.z; dst[3] = x.w;
            }
        }
        if (lane < NS) fv[lane] = fp[lane];
        __syncthreads();

        // W = V f + v
        if (lane < NS) {
            float a = vv[lane];
#pragma unroll
            for (int j = 0; j < NS; ++j) a += V[lane * 17 + j] * fv[j];
            Wv[lane] = a;
        }
        __syncthreads();

        // q = c + F^T W
        if (lane < NT) {
            float a = cp[lane];
#pragma unroll
            for (int s = 0; s < NS; ++s) a += F[s * 17 + lane] * Wv[s];
            qv[lane] = a;
        }

        // M = V * F   (full-wave WMMA)
        v8f acc = v8zero();
        acc = mm16(V, F, acc, m, hi, false);
        storeCD(M, acc, m, hi);
        __syncthreads();

        // Q = F^T * M + C  (C loaded straight into WMMA C layout, coalesced rows)
        v8f qc;
        {
            const int mb = hi * 8;
#pragma unroll
            for (int r = 0; r < 8; ++r) qc[r] = Cp[(mb + r) * 16 + m];
        }
        qc = mm16(F, M, qc, m, hi, true);
        storeCD(Q, qc, m, hi);
        __syncthreads();

        // Solve Quu * X = [Qux | qu]; K = -X[:, :12], k = -X[:, 12].
        // Quu is SPD (C PSD + I) -> pivot-free Gaussian elimination; 13 RHS
        // columns handled by lanes 0..12.
        if (lane < 13) {
            float a[4][4], r4[4], sol[4];
#pragma unroll
            for (int u = 0; u < 4; ++u)
#pragma unroll
                for (int vvv = 0; vvv < 4; ++vvv)
                    a[u][vvv] = Q[(12 + u) * 17 + 12 + vvv];
#pragma unroll
            for (int u = 0; u < 4; ++u)
                r4[u] = (lane < 12) ? Q[(12 + u) * 17 + lane] : qv[12 + u];
#pragma unroll
            for (int p = 0; p < 4; ++p) {
                const float inv = 1.f / a[p][p];
#pragma unroll
                for (int r = 0; r < 4; ++r) {
                    if (r > p) {
                        const float mult = a[r][p] * inv;
#pragma unroll
                        for (int cc = 0; cc < 4; ++cc)
                            if (cc >= p) a[r][cc] -= mult * a[p][cc];
                        r4[r] -= mult * r4[p];
                    }
                }
            }
#pragma unroll
            for (int p = 3; p >= 0; --p) {
                float s = r4[p];
#pragma unroll
                for (int cc = 0; cc < 4; ++cc)
                    if (cc > p) s -= a[p][cc] * sol[cc];
                sol[p] = s / a[p][p];
            }
            if (lane < 12) {
#pragma unroll
                for (int u = 0; u < 4; ++u) KK[u * 12 + lane] = -sol[u];
            } else {
#pragma unroll
                for (int u = 0; u < 4; ++u) KK[48 + u] = -sol[u];
            }
        }
        __syncthreads();

        // Persist K,k for the forward rollout (same wave reads them back).
        {
            float4* kkp4 = (float4*)(Kk + tb * 52);
            for (int i = lane; i < 13; i += 32) {
                float4 x;
                x.x = KK[i * 4 + 0]; x.y = KK[i * 4 + 1];
                x.z = KK[i * 4 + 2]; x.w = KK[i * 4 + 3];
                kkp4[i] = x;
            }
        }

        // Vn = Qxx + Qxu K + K^T Qux + K^T Quu K  (unsymmetrized, into M)
        for (int idx = lane; idx < NS * NS; idx += 32) {
            const int i = idx / NS, j = idx % NS;
            float a = Q[i * 17 + j];
#pragma unroll
            for (int u = 0; u < 4; ++u) {
                float quuK = 0.f;
#pragma unroll
                for (int vvv = 0; vvv < 4; ++vvv)
                    quuK += Q[(12 + u) * 17 + 12 + vvv] * KK[vvv * 12 + j];
                a += Q[i * 17 + 12 + u] * KK[u * 12 + j];
                a += KK[u * 12 + i] * (Q[(12 + u) * 17 + j] + quuK);
            }
            M[i * 17 + j] = a;
        }
        __syncthreads();

        // V = 0.5 (Vn + Vn^T); vn = qx + Qxu k + K^T (qu + Quu k)
        for (int idx = lane; idx < NS * NS; idx += 32) {
            const int i = idx / NS, j = idx % NS;
            V[i * 17 + j] = 0.5f * (M[i * 17 + j] + M[j * 17 + i]);
        }
        if (lane < NS) {
            float a = qv[lane];
#pragma unroll
            for (int u = 0; u < 4; ++u) {
                float tqu = qv[12 + u];
#pragma unroll
                for (int vvv = 0; vvv < 4; ++vvv)
                    tqu += Q[(12 + u) * 17 + 12 + vvv] * KK[48 + vvv];
                a += Q[lane * 17 + 12 + u] * KK[48 + u] + KK[u * 12 + lane] * tqu;
            }
            vv[lane] = a;
        }
        __syncthreads();
    }

    // ---------------- Forward rollout + cost ----------------
    float cost = 0.f;
    if (lane < NS) xv[lane] = x0g[(size_t)b * NS + lane];
    __syncthreads();

    for (int t = 0; t < TS; ++t) {
        const size_t tb = (size_t)t * BB + b;
        const float* Fp = Fg + tb * (NS * NT);
        const float* fp = fg + tb * NS;
        const float* Cp = Cg + tb * (NT * NT);
        const float* cp = cg + tb * NT;
        const float* kkp = Kk + tb * 52;

        // Prefetch next step's C/F tiles.
        if (t < TS - 1) {
            const char* pfa = (lane < 8)
                ? ((const char*)Cp + (size_t)BB * sizeof(float) * NT * NT +
                   (size_t)lane * 128)
                : ((const char*)Fp + (size_t)BB * sizeof(float) * NS * NT +
                   (size_t)(lane - 8) * 128);
            if (lane < 14) __builtin_prefetch(pfa, 0, 3);
        }

        {   // K,k: 13 x b128
            const float4* kkp4 = (const float4*)kkp;
            for (int i = lane; i < 13; i += 32) {
                const float4 x = kkp4[i];
                KK[i * 4 + 0] = x.x; KK[i * 4 + 1] = x.y;
                KK[i * 4 + 2] = x.z; KK[i * 4 + 3] = x.w;
            }
        }
        {   // F tile: b128
            const float4* Fp4 = (const float4*)Fp;
            for (int i = lane; i < (NS * NT) / 4; i += 32) {
                const float4 x = Fp4[i];
                float* dst = &F[(i / 4) * 17 + (i % 4) * 4];
                dst[0] = x.x; dst[1] = x.y; dst[2] = x.z; dst[3] = x.w;
            }
        }
        {   // C tile staging into sQ: b128
            const float4* Cp4 = (const float4*)Cp;
            for (int i = lane; i < (NT * NT) / 4; i += 32) {
                const float4 x = Cp4[i];
                float* dst = &Q[(i / 4) * 17 + (i % 4) * 4];
                dst[0] = x.x; dst[1] = x.y; dst[2] = x.z; dst[3] = x.w;
            }
        }
        __syncthreads();

        // u = K x + k
        if (lane < NC) {
            float a = KK[48 + lane];
#pragma unroll
            for (int s = 0; s < NS; ++s) a += KK[lane * 12 + s] * xv[s];
            xv[12 + lane] = a;
        }
        __syncthreads();

        // outputs
        if (lane < NS) xs[tb * NS + lane] = xv[lane];
        if (lane < NC) us[tb * NC + lane] = xv[12 + lane];

        // cost: sum_i tau_i * (c_i + 0.5 * sum_j C_ij tau_j)
        float part = 0.f;
        if (lane < NT) {
            float h = 0.f;
#pragma unroll
            for (int j = 0; j < NT; ++j) h += Q[lane * 17 + j] * xv[j];
            part = xv[lane] * (cp[lane] + 0.5f * h);
        }
#pragma unroll
        for (int off = 16; off > 0; off >>= 1) part += __shfl_xor(part, off, 32);
        cost += part;

        // x_next = F tau + f
        float xn = 0.f;
        if (lane < NS) {
            xn = fp[lane];
#pragma unroll
            for (int tt = 0; tt < NT; ++tt) xn += F[lane * 17 + tt] * xv[tt];
        }
        __syncthreads();
        if (lane < NS) xv[lane] = xn;
        __syncthreads();
    }

    if (lane == 0) costs[b] = cost;
}

extern "C" void kernel_launch(void* const* d_in, const int* in_sizes, int n_in,
                              void* d_out, int out_size, void* d_ws, size_t ws_size,
                              hipStream_t stream) {
    const float* C  = (const float*)d_in[0];
    const float* c  = (const float*)d_in[1];
    const float* F  = (const float*)d_in[2];
    const float* f  = (const float*)d_in[3];
    const float* x0 = (const float*)d_in[4];
    float* xs    = (float*)d_out;
    float* us    = xs + (size_t)TS * BB * NS;
    float* costs = us + (size_t)TS * BB * NC;
    float* Kk    = (float*)d_ws;  // 128*2048*52 floats = 54.5 MB of scratch
    (void)in_sizes; (void)n_in; (void)out_size; (void)ws_size;
    lqr_kernel<<<dim3(BB / NW), dim3(NW * 32), 0, stream>>>(
        C, c, F, f, x0, xs, us, costs, Kk);
}